// GraphLevelGAT_58171037057467
// MI455X (gfx1250) — compile-verified
//
#include <hip/hip_runtime.h>
#include <hip/hip_bf16.h>

#define N_NODES 100000
#define N_EDGES 1600000
#define E_TOT   1700000   // edges + self loops
#define IN_DIM  128
#define HID     64
#define N_GRAPHS 512
#define NEG_SLOPE 0.2f

typedef float v2f __attribute__((ext_vector_type(2)));
typedef float v8f __attribute__((ext_vector_type(8)));

// ---------------- WMMA f32 GEMM:  H[N,64] = X[N,Fin] @ W[Fin,64] ----------------
// grid.x = N/16 (row tiles), block = 128 threads = 4 waves; wave w owns output
// columns [16w, 16w+16). K-loop issues V_WMMA_F32_16X16X4_F32 per 4 K-values.
// A 16x4 layout: lanes0-15 M=0..15 {K=k0,k0+1}, lanes16-31 {K=k0+2,k0+3}.
// B 4x16 layout: VGPR r holds row K=k0+r+2*half, N = lane&15.
// C/D 16x16:     c[r] is row (r + 8*half), col lane&15.
__global__ __launch_bounds__(128)
void gat_gemm_wmma(const float* __restrict__ X, const float* __restrict__ W,
                   float* __restrict__ H, int Fin) {
  const int lane = threadIdx.x & 31;
  const int wave = threadIdx.x >> 5;
  const int half = lane >> 4;          // 0 or 1
  const int l16  = lane & 15;
  const int mbase = blockIdx.x * 16;
  const int col   = wave * 16 + l16;   // output column in [0,64)

  v8f c = {0.f, 0.f, 0.f, 0.f, 0.f, 0.f, 0.f, 0.f};
  for (int k0 = 0; k0 < Fin; k0 += 4) {
    v2f a, b;
    const float* ap = X + (size_t)(mbase + l16) * Fin + (k0 + 2 * half);
    a.x = ap[0];
    a.y = ap[1];
    const float* bp = W + (size_t)(k0 + 2 * half) * HID + col;
    b.x = bp[0];
    b.y = bp[HID];
    c = __builtin_amdgcn_wmma_f32_16x16x4_f32(false, a, false, b,
                                              (short)0, c, false, false);
  }
#pragma unroll
  for (int r = 0; r < 8; ++r)
    H[(size_t)(mbase + r + 8 * half) * HID + col] = c[r];
}

// ---------------- per-node attention logit halves ----------------
__global__ void node_dots(const float* __restrict__ h,
                          const float* __restrict__ a_src,
                          const float* __restrict__ a_dst,
                          float* __restrict__ as, float* __restrict__ ad) {
  int i = blockIdx.x * blockDim.x + threadIdx.x;
  if (i >= N_NODES) return;
  const float* hp = h + (size_t)i * HID;
  float s0 = 0.f, s1 = 0.f;
#pragma unroll 8
  for (int k = 0; k < HID; ++k) {
    float v = hp[k];
    s0 += v * a_src[k];
    s1 += v * a_dst[k];
  }
  as[i] = s0;
  ad[i] = s1;
}

// ---------------- order-preserving float<->uint for atomic max ----------------
__device__ __forceinline__ unsigned f2ord(float f) {
  unsigned u = __float_as_uint(f);
  return (u & 0x80000000u) ? ~u : (u | 0x80000000u);
}
__device__ __forceinline__ float ord2f(unsigned u) {
  return __uint_as_float((u & 0x80000000u) ? (u & 0x7FFFFFFFu) : ~u);
}

__device__ __forceinline__ void edge_sd(const long long* __restrict__ src,
                                        const long long* __restrict__ dst,
                                        int e, int& s, int& d) {
  if (e < N_EDGES) { s = (int)src[e]; d = (int)dst[e]; }
  else             { s = d = e - N_EDGES; }          // self loops
}

// pass 1: leaky-relu logit, store it, atomic running max per dst
__global__ void edge_logits(const long long* __restrict__ src,
                            const long long* __restrict__ dst,
                            const float* __restrict__ as,
                            const float* __restrict__ ad,
                            float* __restrict__ logit,
                            unsigned* __restrict__ mOrd) {
  int e = blockIdx.x * blockDim.x + threadIdx.x;
  if (e >= E_TOT) return;
  int s, d; edge_sd(src, dst, e, s, d);
  float l = as[s] + ad[d];
  l = (l > 0.f) ? l : NEG_SLOPE * l;
  logit[e] = l;
  atomicMax(&mOrd[d], f2ord(l));
}

// pass 2: p = exp(l - max), in-place over logit buffer, atomic denom per dst
__global__ void edge_exp(const long long* __restrict__ src,
                         const long long* __restrict__ dst,
                         float* __restrict__ logit,
                         const unsigned* __restrict__ mOrd,
                         float* __restrict__ denom) {
  int e = blockIdx.x * blockDim.x + threadIdx.x;
  if (e >= E_TOT) return;
  int s, d; edge_sd(src, dst, e, s, d);
  float p = __expf(logit[e] - ord2f(mOrd[d]));
  logit[e] = p;
  atomicAdd(&denom[d], p);
}

// pass 3: acc[dst] += p * h[src]; one wave per edge, float2 per lane (64 ch)
__global__ void edge_scatter(const long long* __restrict__ src,
                             const long long* __restrict__ dst,
                             const float* __restrict__ p,
                             const float* __restrict__ h,
                             float* __restrict__ acc) {
  long long gid = (long long)blockIdx.x * blockDim.x + threadIdx.x;
  int e = (int)(gid >> 5);
  int lane = (int)(gid & 31);
  if (e >= E_TOT) return;
  int s, d; edge_sd(src, dst, e, s, d);
  float pe = p[e];
  float2 hv = reinterpret_cast<const float2*>(h + (size_t)s * HID)[lane];
  float* ap = acc + (size_t)d * HID + 2 * lane;
  atomicAdd(ap + 0, pe * hv.x);
  atomicAdd(ap + 1, pe * hv.y);
}

// epilogue: normalize by denom, add bias, optional relu (in place)
__global__ void node_finish(float* __restrict__ acc,
                            const float* __restrict__ denom,
                            const float* __restrict__ bias, int do_relu) {
  int t = blockIdx.x * blockDim.x + threadIdx.x;
  if (t >= N_NODES * HID) return;
  int i = t >> 6, c = t & 63;
  float v = acc[t] / denom[i] + bias[c];
  if (do_relu) v = fmaxf(v, 0.f);
  acc[t] = v;
}

// ---------------- pooling ----------------
__global__ void pool_cnt(const long long* __restrict__ batch, float* __restrict__ cnt) {
  int i = blockIdx.x * blockDim.x + threadIdx.x;
  if (i >= N_NODES) return;
  atomicAdd(&cnt[(int)batch[i]], 1.0f);
}
__global__ void pool_sum(const float* __restrict__ h,
                         const long long* __restrict__ batch,
                         float* __restrict__ sums) {
  int t = blockIdx.x * blockDim.x + threadIdx.x;
  if (t >= N_NODES * HID) return;
  int i = t >> 6, c = t & 63;
  atomicAdd(&sums[(int)batch[i] * HID + c], h[t]);
}
__global__ void pool_div(const float* __restrict__ sums,
                         const float* __restrict__ cnt, float* __restrict__ out) {
  int t = blockIdx.x * blockDim.x + threadIdx.x;
  if (t >= N_GRAPHS * HID) return;
  out[t] = sums[t] / fmaxf(cnt[t >> 6], 1.0f);
}

__global__ void zero_f32(float* __restrict__ p, int n) {
  int i = blockIdx.x * blockDim.x + threadIdx.x;
  if (i < n) p[i] = 0.f;
}

static inline int cdiv(long long a, long long b) { return (int)((a + b - 1) / b); }

extern "C" void kernel_launch(void* const* d_in, const int* in_sizes, int n_in,
                              void* d_out, int out_size, void* d_ws, size_t ws_size,
                              hipStream_t stream) {
  const float*     x      = (const float*)d_in[0];
  const long long* ei     = (const long long*)d_in[1];   // [2, E] int64
  const long long* batch  = (const long long*)d_in[2];
  const float*     W1     = (const float*)d_in[3];
  const float*     asrc1  = (const float*)d_in[4];
  const float*     adst1  = (const float*)d_in[5];
  const float*     b1     = (const float*)d_in[6];
  const float*     W2     = (const float*)d_in[7];
  const float*     asrc2  = (const float*)d_in[8];
  const float*     adst2  = (const float*)d_in[9];
  const float*     b2     = (const float*)d_in[10];
  float*           out    = (float*)d_out;

  const long long* srcIdx = ei;
  const long long* dstIdx = ei + N_EDGES;

  const size_t NH = (size_t)N_NODES * HID;
  float* ws    = (float*)d_ws;
  float* bufA  = ws;                 // h1  (layer1 GEMM out), later acc2
  float* bufB  = bufA + NH;          // acc1, later layer-2 input g
  float* bufC  = bufB + NH;          // h2  (layer2 GEMM out)
  float* asv   = bufC + NH;          // [N]
  float* adv   = asv  + N_NODES;     // [N]
  float* mBuf  = adv  + N_NODES;     // [N]  (unsigned ordmax, bits)
  float* denom = mBuf + N_NODES;     // [N]
  float* pBuf  = denom + N_NODES;    // [E_TOT] logits then p
  float* sums  = pBuf + E_TOT;       // [512*64]
  float* cnt   = sums + N_GRAPHS * HID; // [512]

  const int TB = 256;
  const int gN   = cdiv(N_NODES, TB);
  const int gNH  = cdiv((long long)N_NODES * HID, TB);
  const int gE   = cdiv(E_TOT, TB);
  const int gE32 = cdiv((long long)E_TOT * 32, TB);

  // ---------------- layer 1 ----------------
  gat_gemm_wmma<<<N_NODES / 16, 128, 0, stream>>>(x, W1, bufA, IN_DIM);
  node_dots<<<gN, TB, 0, stream>>>(bufA, asrc1, adst1, asv, adv);
  zero_f32<<<gN, TB, 0, stream>>>(mBuf, N_NODES);     // ord(-NaN) floor
  zero_f32<<<gN, TB, 0, stream>>>(denom, N_NODES);
  zero_f32<<<gNH, TB, 0, stream>>>(bufB, (int)NH);
  edge_logits<<<gE, TB, 0, stream>>>(srcIdx, dstIdx, asv, adv, pBuf, (unsigned*)mBuf);
  edge_exp<<<gE, TB, 0, stream>>>(srcIdx, dstIdx, pBuf, (const unsigned*)mBuf, denom);
  edge_scatter<<<gE32, TB, 0, stream>>>(srcIdx, dstIdx, pBuf, bufA, bufB);
  node_finish<<<gNH, TB, 0, stream>>>(bufB, denom, b1, /*relu=*/1);

  // ---------------- layer 2 ----------------
  gat_gemm_wmma<<<N_NODES / 16, 128, 0, stream>>>(bufB, W2, bufC, HID);
  node_dots<<<gN, TB, 0, stream>>>(bufC, asrc2, adst2, asv, adv);
  zero_f32<<<gN, TB, 0, stream>>>(mBuf, N_NODES);
  zero_f32<<<gN, TB, 0, stream>>>(denom, N_NODES);
  zero_f32<<<gNH, TB, 0, stream>>>(bufA, (int)NH);
  edge_logits<<<gE, TB, 0, stream>>>(srcIdx, dstIdx, asv, adv, pBuf, (unsigned*)mBuf);
  edge_exp<<<gE, TB, 0, stream>>>(srcIdx, dstIdx, pBuf, (const unsigned*)mBuf, denom);
  edge_scatter<<<gE32, TB, 0, stream>>>(srcIdx, dstIdx, pBuf, bufC, bufA);
  node_finish<<<gNH, TB, 0, stream>>>(bufA, denom, b2, /*relu=*/0);

  // ---------------- global mean pool ----------------
  zero_f32<<<cdiv(N_GRAPHS * HID + N_GRAPHS, TB), TB, 0, stream>>>(sums, N_GRAPHS * HID + N_GRAPHS);
  pool_cnt<<<gN, TB, 0, stream>>>(batch, cnt);
  pool_sum<<<gNH, TB, 0, stream>>>(bufA, batch, sums);
  pool_div<<<cdiv(N_GRAPHS * HID, TB), TB, 0, stream>>>(sums, cnt, out);
}